// TropicalMaxReLu_70832600646268
// MI455X (gfx1250) — compile-verified
//
#include <hip/hip_runtime.h>

// Tropical (max-plus) matmul with ReLU->-inf masking on X.
// out[n,o] = max_i ( (X[n,i] > 0 ? -inf : X[n,i]) + W[o,i] )
// N=4096, D_IN=1024, D_OUT=1024, fp32.
//
// MI455X / gfx1250 (wave32) strategy:
//  - Max-plus has no WMMA mapping; this is a pure-VALU-throughput problem
//    (8.6 G add/max ops vs ~36 MB of traffic => compute-bound by ~50x).
//  - 128x128 output tile per 256-thread block, 8x8 register tile per thread.
//  - K chunks of 32 staged in LDS, double buffered, filled with
//    global_load_async_to_lds_b128 (ASYNCcnt path).
//  - Inner loop forced via inline asm to v_pk_add_f32 + v_max3_num_f32:
//    exactly 1 VALU instruction per (add,max) element pair.

#define N_ROWS 4096
#define D_INF  1024   // inner dim
#define D_OUTF 1024
#define TILE_M 128
#define TILE_O 128
#define TILE_K 32
#define K_PAD  36     // +4 floats: rows stay 16B-aligned (144B) and bank-conflict-free

typedef float v2f __attribute__((ext_vector_type(2)));
typedef float v4f __attribute__((ext_vector_type(4)));
typedef int   v4i __attribute__((vector_size(16)));   // b128 payload type for async builtin

#define AS1 __attribute__((address_space(1)))
#define AS3 __attribute__((address_space(3)))

__device__ __forceinline__ void async_copy_b128(const float* g, float* l) {
#if __has_builtin(__builtin_amdgcn_global_load_async_to_lds_b128)
  __builtin_amdgcn_global_load_async_to_lds_b128(
      (AS1 v4i*)g, (AS3 v4i*)l, /*offset=*/0, /*cpol=*/0);
#else
  unsigned lds_addr = (unsigned)(unsigned long long)l;  // low 32 bits = LDS offset
  asm volatile("global_load_async_to_lds_b128 %0, %1, off"
               :: "v"(lds_addr), "v"(g)
               : "memory");
#endif
}

__device__ __forceinline__ void wait_async0() {
#if __has_builtin(__builtin_amdgcn_s_wait_asynccnt)
  __builtin_amdgcn_s_wait_asynccnt(0);
#else
  asm volatile("s_wait_asynccnt 0" ::: "memory");
#endif
}

// acc = max(acc, a.x + b.x, a.y + b.y) in exactly two instruction slots:
//   v_pk_add_f32  (2 adds)  +  v_max3_num_f32 (2 maxes)
__device__ __forceinline__ void mp_update(float& acc, v2f a, v2f b) {
  v2f s;
  asm("v_pk_add_f32 %0, %1, %2" : "=v"(s) : "v"(a), "v"(b));
  asm("v_max3_num_f32 %0, %1, %2, %0" : "+v"(acc) : "v"(s.x), "v"(s.y));
}

__global__ __launch_bounds__(256)
void tropical_maxplus_kernel(const float* __restrict__ X,
                             const float* __restrict__ W,
                             float* __restrict__ out) {
  __shared__ float Xs[2][TILE_M * K_PAD];
  __shared__ float Ws[2][TILE_O * K_PAD];

  const int tid = threadIdx.x;
  const int tx  = tid & 15;   // output-col group (o = block_o + tx + 16*oi)
  const int ty  = tid >> 4;   // row group       (m = block_m + ty + 16*mi)
  const int block_m = blockIdx.y * TILE_M;
  const int block_o = blockIdx.x * TILE_O;

  // Each thread issues 4 b128 copies per tile per matrix:
  // chunk c = tid + 256*i ; row = c>>3 (0..127) ; q = c&7 (16B column slot)
  auto issue_loads = [&](int k0, int b) {
#pragma unroll
    for (int i = 0; i < 4; ++i) {
      const int c   = tid + 256 * i;
      const int row = c >> 3;
      const int q   = c & 7;
      const float* gX = X + (size_t)(block_m + row) * D_INF + k0 + q * 4;
      const float* gW = W + (size_t)(block_o + row) * D_INF + k0 + q * 4;
      async_copy_b128(gX, &Xs[b][row * K_PAD + q * 4]);
      async_copy_b128(gW, &Ws[b][row * K_PAD + q * 4]);
    }
  };

  float acc[8][8];
#pragma unroll
  for (int mi = 0; mi < 8; ++mi)
#pragma unroll
    for (int oi = 0; oi < 8; ++oi)
      acc[mi][oi] = -__builtin_inff();

  issue_loads(0, 0);

  const int NCHUNK = D_INF / TILE_K;  // 32
  for (int t = 0; t < NCHUNK; ++t) {
    const int b = t & 1;
    wait_async0();        // this wave's async copies for chunk t are in LDS
    __syncthreads();      // all waves' copies are in LDS

    // ReLU mask pass: positive X entries -> -inf (max-plus identity).
    // Each of the 1024 16B chunks of the X tile is touched exactly once.
#pragma unroll
    for (int i = 0; i < 4; ++i) {
      const int c   = tid + 256 * i;
      const int row = c >> 3;
      const int q   = c & 7;
      v4f* p = (v4f*)&Xs[b][row * K_PAD + q * 4];
      v4f v = *p;
#pragma unroll
      for (int j = 0; j < 4; ++j)
        v[j] = (v[j] > 0.0f) ? -__builtin_inff() : v[j];
      *p = v;
    }
    __syncthreads();

    if (t + 1 < NCHUNK) issue_loads((t + 1) * TILE_K, b ^ 1);

    // Core loop: 4 k-values per iteration.
    //   16 ds_load_b128 fragment loads, then per accumulator:
    //   2x (v_pk_add_f32 + v_max3_num_f32) = 256 VALU slots per 256 updates.
#pragma unroll 2
    for (int kk = 0; kk < TILE_K; kk += 4) {
      v4f xq[8], wq[8];
#pragma unroll
      for (int mi = 0; mi < 8; ++mi)
        xq[mi] = *(const v4f*)&Xs[b][(ty + 16 * mi) * K_PAD + kk];
#pragma unroll
      for (int oi = 0; oi < 8; ++oi)
        wq[oi] = *(const v4f*)&Ws[b][(tx + 16 * oi) * K_PAD + kk];
#pragma unroll
      for (int mi = 0; mi < 8; ++mi)
#pragma unroll
        for (int oi = 0; oi < 8; ++oi) {
          mp_update(acc[mi][oi], xq[mi].xy, wq[oi].xy);
          mp_update(acc[mi][oi], xq[mi].zw, wq[oi].zw);
        }
    }
  }

  // Store 8x8 tile; lanes tx=0..15 write 16 consecutive floats (coalesced).
#pragma unroll
  for (int mi = 0; mi < 8; ++mi) {
    const size_t row = (size_t)(block_m + ty + 16 * mi);
#pragma unroll
    for (int oi = 0; oi < 8; ++oi)
      out[row * D_OUTF + block_o + tx + 16 * oi] = acc[mi][oi];
  }
}

extern "C" void kernel_launch(void* const* d_in, const int* in_sizes, int n_in,
                              void* d_out, int out_size, void* d_ws, size_t ws_size,
                              hipStream_t stream) {
  (void)in_sizes; (void)n_in; (void)d_ws; (void)ws_size; (void)out_size;
  const float* X = (const float*)d_in[0];   // [N, D_IN]
  const float* W = (const float*)d_in[1];   // [D_OUT, D_IN]
  float* out = (float*)d_out;               // [N, D_OUT]

  dim3 grid(D_OUTF / TILE_O, N_ROWS / TILE_M);  // (8, 32)
  tropical_maxplus_kernel<<<grid, 256, 0, stream>>>(X, W, out);
}